// TransitionDownModule_51651276702288
// MI455X (gfx1250) — compile-verified
//
#include <hip/hip_runtime.h>
#include <hip/hip_bf16.h>

typedef float v2f __attribute__((ext_vector_type(2)));
typedef float v8f __attribute__((ext_vector_type(8)));

#define NB      8
#define NPTS    8192
#define NQ      2048      // n2
#define KNN     16
#define DIN     64
#define DOUT    128
#define NQTOT   (NB*NQ)          // 16384 queries
#define NROWS   (NQTOT*KNN)      // 262144 gathered rows
#define CNT_INV (1.0f/262144.0f)
#define HSTRIDE 132              // padded LDS stride for h staging

static __device__ __forceinline__ v8f wmma4(v2f a, v2f b, v8f c) {
    return __builtin_amdgcn_wmma_f32_16x16x4_f32(false, a, false, b, (short)0, c, false, false);
}

// ---------------------------------------------------------------------------
// K0: repack W2 (128x128) into float2 K-pairs
// ---------------------------------------------------------------------------
__global__ void td_packw2_kernel(const float* __restrict__ W2, v2f* __restrict__ w2p) {
    const int i  = blockIdx.x * 128 + threadIdx.x;   // 0..8191
    const int pr = i >> 7;
    const int col = i & 127;
    v2f t;
    t.x = W2[(2*pr)   * DOUT + col];
    t.y = W2[(2*pr+1) * DOUT + col];
    w2p[i] = t;
}

// ---------------------------------------------------------------------------
// K1: brute-force KNN (top-16 smallest d), write idx + p_out, zero stats
// ---------------------------------------------------------------------------
__global__ void td_knn_kernel(const float* __restrict__ p,
                              int* __restrict__ idxbuf,
                              float* __restrict__ stats,
                              float* __restrict__ p_out) {
    const int g = blockIdx.x * 256 + threadIdx.x;     // query id 0..16383
    const int b = g >> 11;
    const int m = g & 2047;
    if (blockIdx.x == 0) stats[threadIdx.x] = 0.0f;

    const float* pb = p + (size_t)b * NPTS * 3;
    const float qx = pb[(4*m)*3 + 0];
    const float qy = pb[(4*m)*3 + 1];
    const float qz = pb[(4*m)*3 + 2];
    const float q2 = qx*qx + qy*qy + qz*qz;

    float bd[KNN]; int bi[KNN];
#pragma unroll
    for (int k = 0; k < KNN; ++k) { bd[k] = 3.4e38f; bi[k] = 0; }
    float wv = 3.4e38f;

    __shared__ float sP[512*3];
    for (int c0 = 0; c0 < NPTS; c0 += 512) {
        __syncthreads();
        for (int t = threadIdx.x; t < 512*3; t += 256) sP[t] = pb[c0*3 + t];
        __syncthreads();
        for (int jj = 0; jj < 512; ++jj) {
            const float px = sP[jj*3+0], py = sP[jj*3+1], pz = sP[jj*3+2];
            const float dot = qx*px + qy*py + qz*pz;
            const float p2  = px*px + py*py + pz*pz;
            const float d   = q2 + p2 - 2.0f*dot;
            if (d < wv) {
                bool done = false;
#pragma unroll
                for (int k = 0; k < KNN; ++k) {
                    if (!done && bd[k] == wv) { bd[k] = d; bi[k] = c0 + jj; done = true; }
                }
                wv = bd[0];
#pragma unroll
                for (int k = 1; k < KNN; ++k) wv = fmaxf(wv, bd[k]);
            }
        }
    }
#pragma unroll
    for (int k = 0; k < KNN; ++k) idxbuf[(size_t)g*KNN + k] = bi[k];
    p_out[(size_t)g*3 + 0] = qx;
    p_out[(size_t)g*3 + 1] = qy;
    p_out[(size_t)g*3 + 2] = qz;
}

// cooperative: pack W1 into LDS as float2 K-pairs
static __device__ __forceinline__ void load_w1_packed(const float* __restrict__ W1,
                                                      v2f* __restrict__ sW1p,
                                                      int tid, int nthreads) {
    for (int i = tid; i < 32*DOUT; i += nthreads) {
        const int pr  = i >> 7;
        const int col = i & 127;
        v2f t;
        t.x = W1[(2*pr)   * DOUT + col];
        t.y = W1[(2*pr+1) * DOUT + col];
        sW1p[i] = t;
    }
}

// GEMM1 for one 16-row tile; rolled K-loop (2 K-pairs per iter) so that
// B-fragment loads cannot be hoisted across tiles (avoids scratch spills).
static __device__ __forceinline__ void gemm1_tile(const float* __restrict__ xrow,
                                                  const v2f* __restrict__ sW1p,
                                                  int laneM, int hiHalf, int koff,
                                                  v8f acc[8]) {
#pragma unroll 1
    for (int kk = 0; kk < 16; kk += 2) {
        const v2f a0 = *(const v2f*)(xrow + 4*kk + koff);
        const v2f a1 = *(const v2f*)(xrow + 4*kk + 4 + koff);
        const v2f* wp0 = sW1p + (2*kk     + hiHalf) * DOUT + laneM;
        const v2f* wp1 = sW1p + (2*kk + 2 + hiHalf) * DOUT + laneM;
        v2f b0[8], b1[8];
#pragma unroll
        for (int n = 0; n < 8; ++n) b0[n] = wp0[16*n];
#pragma unroll
        for (int n = 0; n < 8; ++n) b1[n] = wp1[16*n];
#pragma unroll
        for (int n = 0; n < 8; ++n) acc[n] = wmma4(a0, b0[n], acc[n]);
#pragma unroll
        for (int n = 0; n < 8; ++n) acc[n] = wmma4(a1, b1[n], acc[n]);
    }
}

// ---------------------------------------------------------------------------
// K2: gather + GEMM1 via f32 WMMA, accumulate per-channel sum / sumsq
// ---------------------------------------------------------------------------
__global__ void td_stats_kernel(const float* __restrict__ x,
                                const float* __restrict__ W1,
                                const int*   __restrict__ idxbuf,
                                float* __restrict__ gSum,
                                float* __restrict__ gSq) {
    __shared__ v2f   sW1p[32*DOUT];   // 32 KB, K-pair packed
    __shared__ float sSum[DOUT];
    __shared__ float sSq[DOUT];

    load_w1_packed(W1, sW1p, threadIdx.x, 128);
    if (threadIdx.x < DOUT) { sSum[threadIdx.x] = 0.0f; sSq[threadIdx.x] = 0.0f; }
    __syncthreads();

    const int lane   = threadIdx.x & 31;
    const int wid    = threadIdx.x >> 5;
    const int laneM  = lane & 15;
    const int hiHalf = lane >> 4;
    const int koff   = hiHalf * 2;

#pragma unroll 1
    for (int t = 0; t < 4; ++t) {
        const int tile = blockIdx.x * 16 + wid * 4 + t;
        const int s    = tile * 16 + laneM;
        const int j    = idxbuf[s];
        const int b    = s >> 15;
        const float* xrow = x + ((size_t)((b << 13) + j) << 6);

        v8f acc[8];
#pragma unroll
        for (int n = 0; n < 8; ++n) acc[n] = (v8f){0,0,0,0,0,0,0,0};
        gemm1_tile(xrow, sW1p, laneM, hiHalf, koff, acc);

#pragma unroll
        for (int n = 0; n < 8; ++n) {
            float s1 = 0.0f, s2 = 0.0f;
#pragma unroll
            for (int r = 0; r < 8; ++r) { const float v = acc[n][r]; s1 += v; s2 += v*v; }
            atomicAdd(&sSum[16*n + laneM], s1);
            atomicAdd(&sSq [16*n + laneM], s2);
        }
    }
    __syncthreads();
    if (threadIdx.x < DOUT) {
        atomicAdd(&gSum[threadIdx.x], sSum[threadIdx.x]);
        atomicAdd(&gSq [threadIdx.x], sSq [threadIdx.x]);
    }
}

// ---------------------------------------------------------------------------
// K3: finalize BN coefficients
// ---------------------------------------------------------------------------
__global__ void td_bn_kernel(const float* __restrict__ gSum,
                             const float* __restrict__ gSq,
                             const float* __restrict__ gamma,
                             const float* __restrict__ beta,
                             float* __restrict__ gA,
                             float* __restrict__ gC) {
    const int d = threadIdx.x;
    if (d < DOUT) {
        const float mean = gSum[d] * CNT_INV;
        const float var  = gSq[d] * CNT_INV - mean*mean;
        const float a    = gamma[d] * rsqrtf(var + 1e-5f);
        gA[d] = a;
        gC[d] = beta[d] - mean * a;
    }
}

// ---------------------------------------------------------------------------
// K4: GEMM1 (WMMA) -> BN+ReLU -> LDS re-stage -> GEMM2 (WMMA) -> max_k -> +b2
// ---------------------------------------------------------------------------
__global__ void td_main_kernel(const float* __restrict__ x,
                               const float* __restrict__ W1,
                               const v2f*   __restrict__ w2p,
                               const float* __restrict__ b2,
                               const int*   __restrict__ idxbuf,
                               const float* __restrict__ gA,
                               const float* __restrict__ gC,
                               float* __restrict__ out) {
    __shared__ v2f   sW1p[32*DOUT];          // 32 KB
    __shared__ float sH[2][16*HSTRIDE];      // 2 x 8448 B

    load_w1_packed(W1, sW1p, threadIdx.x, 64);
    __syncthreads();

    const int lane   = threadIdx.x & 31;
    const int wid    = threadIdx.x >> 5;
    const int laneM  = lane & 15;
    const int hiHalf = lane >> 4;
    const int koff   = hiHalf * 2;
    float* hst = &sH[wid][0];

    float aC[8], cC[8], b2v[8];
#pragma unroll
    for (int n = 0; n < 8; ++n) {
        const int col = 16*n + laneM;
        aC[n] = gA[col]; cC[n] = gC[col]; b2v[n] = b2[col];
    }

#pragma unroll 1
    for (int t = 0; t < 4; ++t) {
        const int tile = blockIdx.x * 8 + wid * 4 + t;   // == query index
        const int s    = tile * 16 + laneM;
        const int j    = idxbuf[s];
        const int b    = s >> 15;
        const float* xrow = x + ((size_t)((b << 13) + j) << 6);

        // ---- GEMM1: h = x2 @ W1 ----
        v8f acc[8];
#pragma unroll
        for (int n = 0; n < 8; ++n) acc[n] = (v8f){0,0,0,0,0,0,0,0};
        gemm1_tile(xrow, sW1p, laneM, hiHalf, koff, acc);

        // ---- BN + ReLU, stage 16x128 tile to LDS (padded stride) ----
#pragma unroll
        for (int n = 0; n < 8; ++n) {
            const int col = 16*n + laneM;
#pragma unroll
            for (int r = 0; r < 8; ++r) {
                const int M = r + hiHalf*8;
                float v = acc[n][r] * aC[n] + cC[n];
                v = fmaxf(v, 0.0f);
                hst[M*HSTRIDE + col] = v;
            }
        }
        asm volatile("s_wait_dscnt 0" ::: "memory");

        // ---- GEMM2: g = relu_h @ W2 ----  (rolled K-loop, 2 pairs/iter)
        v8f acc2[8];
#pragma unroll
        for (int n = 0; n < 8; ++n) acc2[n] = (v8f){0,0,0,0,0,0,0,0};
#pragma unroll 1
        for (int kk = 0; kk < 32; kk += 2) {
            const v2f a0 = *(const v2f*)(hst + laneM*HSTRIDE + 4*kk + koff);
            const v2f a1 = *(const v2f*)(hst + laneM*HSTRIDE + 4*kk + 4 + koff);
            const v2f* wp0 = w2p + (2*kk     + hiHalf) * DOUT + laneM;
            const v2f* wp1 = w2p + (2*kk + 2 + hiHalf) * DOUT + laneM;
            v2f b0[8], b1[8];
#pragma unroll
            for (int n = 0; n < 8; ++n) b0[n] = wp0[16*n];
#pragma unroll
            for (int n = 0; n < 8; ++n) b1[n] = wp1[16*n];
#pragma unroll
            for (int n = 0; n < 8; ++n) acc2[n] = wmma4(a0, b0[n], acc2[n]);
#pragma unroll
            for (int n = 0; n < 8; ++n) acc2[n] = wmma4(a1, b1[n], acc2[n]);
        }

        // ---- max over the 16 K-rows (one query) + b2, store ----
#pragma unroll
        for (int n = 0; n < 8; ++n) {
            float mx = acc2[n][0];
#pragma unroll
            for (int r = 1; r < 8; ++r) mx = fmaxf(mx, acc2[n][r]);
            mx = fmaxf(mx, __shfl_xor(mx, 16, 32));
            if (hiHalf == 0) {
                const int col = 16*n + laneM;
                out[(size_t)tile * DOUT + col] = mx + b2v[n];
            }
        }
    }
}

// ---------------------------------------------------------------------------
extern "C" void kernel_launch(void* const* d_in, const int* in_sizes, int n_in,
                              void* d_out, int out_size, void* d_ws, size_t ws_size,
                              hipStream_t stream) {
    const float* x     = (const float*)d_in[0];
    const float* p     = (const float*)d_in[1];
    // d_in[2] = n2 (hardcoded 2048)
    const float* W1    = (const float*)d_in[3];
    // d_in[4] = b1 (cancels exactly in BatchNorm)
    const float* gamma = (const float*)d_in[5];
    const float* beta  = (const float*)d_in[6];
    const float* W2    = (const float*)d_in[7];
    const float* b2    = (const float*)d_in[8];

    float* out   = (float*)d_out;                       // [16384 x 128]
    float* p_out = out + (size_t)NQTOT * DOUT;          // [16384 x 3]

    float* ws   = (float*)d_ws;
    float* gSum = ws;             // 128
    float* gSq  = ws + 128;       // 128
    float* gA   = ws + 256;       // 128
    float* gC   = ws + 384;       // 128
    int*   idxb = (int*)(ws + 512);                     // 262144 ints
    v2f*   w2p  = (v2f*)(ws + 512 + NROWS);             // 8192 float2 (64 KB)

    td_packw2_kernel<<<  64, 128, 0, stream>>>(W2, w2p);
    td_knn_kernel   <<<  64, 256, 0, stream>>>(p, idxb, ws, p_out);
    td_stats_kernel <<<1024, 128, 0, stream>>>(x, W1, idxb, gSum, gSq);
    td_bn_kernel    <<<   1, 128, 0, stream>>>(gSum, gSq, gamma, beta, gA, gC);
    td_main_kernel  <<<2048,  64, 0, stream>>>(x, W1, w2p, b2, idxb, gA, gC, out);
}